// VQVAE2_68874095558704
// MI455X (gfx1250) — compile-verified
//
#include <hip/hip_runtime.h>

// VQ-VAE forward for MI455X (gfx1250, wave32).
// Compute-bound (~180 GFLOP, AI ~900 FLOP/B): all matmuls run on the CDNA5
// matrix pipe via V_WMMA_F32_16X16X4_F32 (full f32 fidelity -> argmin-safe).
// GEMM: 8-wave blocks, 256(M)x64(N) block tile, 32(M)x64(N) per wave
// (2 A-frags x 4 B-frags -> 8 WMMAs per k-step, 1.25 mem-instr/WMMA).

typedef __attribute__((ext_vector_type(2))) float v2f;
typedef __attribute__((ext_vector_type(8))) float v8f;

__device__ __forceinline__ v8f wmma4(v2f a, v2f b, v8f c) {
  return __builtin_amdgcn_wmma_f32_16x16x4_f32(false, a, false, b, (short)0, c,
                                               false, false);
}

#define KC 32          // K-chunk staged in LDS
#define LDS_PAD 72     // (2*PAD) mod 64 == 16 -> conflict-free frag reads

// C[M,N] = act(A[M,K] @ W[K,N] + bias[N]); EPI: 0=bias, 1=relu, 2=leaky(0.1)
template <int EPI>
__global__ __launch_bounds__(256) void wmma_gemm_f32(
    const float* __restrict__ A, const float* __restrict__ W,
    const float* __restrict__ bias, float* __restrict__ C, int K, int N) {
  __shared__ float sB[KC][LDS_PAD];

  const int tid  = threadIdx.x;
  const int lane = tid & 31;
  const int wave = tid >> 5;
  const int l15  = lane & 15;
  const int half = lane >> 4;           // 0: K=k,k+1   1: K=k+2,k+3
  const int koff = half * 2;

  const int nBase = blockIdx.x * 64;
  const int mBase = blockIdx.y * 256 + wave * 32;

  const float* __restrict__ Arow0 = A + (size_t)(mBase + l15) * K;
  const float* __restrict__ Arow1 = A + (size_t)(mBase + 16 + l15) * K;

  // staging map: chunk element e = i*256 + tid -> (row = tid/64 + 4i, col = tid%64)
  const int r0 = tid >> 6;              // 0..3
  const int c0 = tid & 63;              // 0..63
  float st[KC / 4];

#pragma unroll
  for (int i = 0; i < KC / 4; ++i)
    st[i] = W[(size_t)(r0 + 4 * i) * N + nBase + c0];

  v8f accA0 = {}, accA1 = {}, accA2 = {}, accA3 = {};
  v8f accB0 = {}, accB1 = {}, accB2 = {}, accB3 = {};

  for (int kb = 0; kb < K; kb += KC) {
#pragma unroll
    for (int i = 0; i < KC / 4; ++i) sB[r0 + 4 * i][c0] = st[i];
    __syncthreads();

    if (kb + KC < K) {
#pragma unroll
      for (int i = 0; i < KC / 4; ++i)
        st[i] = W[(size_t)(kb + KC + r0 + 4 * i) * N + nBase + c0];
    }

#pragma unroll
    for (int kk = 0; kk < KC; kk += 4) {
      const int k0 = kk + koff;
      v2f a0, a1;
      a0.x = Arow0[kb + k0];
      a0.y = Arow0[kb + k0 + 1];
      a1.x = Arow1[kb + k0];
      a1.y = Arow1[kb + k0 + 1];
      v2f b0, b1, b2, b3;
      b0.x = sB[k0][l15];      b0.y = sB[k0 + 1][l15];
      b1.x = sB[k0][l15 + 16]; b1.y = sB[k0 + 1][l15 + 16];
      b2.x = sB[k0][l15 + 32]; b2.y = sB[k0 + 1][l15 + 32];
      b3.x = sB[k0][l15 + 48]; b3.y = sB[k0 + 1][l15 + 48];
      accA0 = wmma4(a0, b0, accA0);
      accA1 = wmma4(a0, b1, accA1);
      accA2 = wmma4(a0, b2, accA2);
      accA3 = wmma4(a0, b3, accA3);
      accB0 = wmma4(a1, b0, accB0);
      accB1 = wmma4(a1, b1, accB1);
      accB2 = wmma4(a1, b2, accB2);
      accB3 = wmma4(a1, b3, accB3);
    }
    __syncthreads();
  }

  // C/D layout: VGPR r -> lanes 0-15: M=r, N=l15 ; lanes 16-31: M=r+8, N=l15
  const int nCol = nBase + l15;
  const float bz0 = bias[nCol], bz1 = bias[nCol + 16];
  const float bz2 = bias[nCol + 32], bz3 = bias[nCol + 48];

#pragma unroll
  for (int sub = 0; sub < 2; ++sub) {
    const int mOut = mBase + sub * 16 + half * 8;
#pragma unroll
    for (int r = 0; r < 8; ++r) {
      float* Crow = C + (size_t)(mOut + r) * N + nCol;
      float v0 = (sub == 0 ? accA0[r] : accB0[r]) + bz0;
      float v1 = (sub == 0 ? accA1[r] : accB1[r]) + bz1;
      float v2 = (sub == 0 ? accA2[r] : accB2[r]) + bz2;
      float v3 = (sub == 0 ? accA3[r] : accB3[r]) + bz3;
      if (EPI == 1) {
        v0 = fmaxf(v0, 0.f); v1 = fmaxf(v1, 0.f);
        v2 = fmaxf(v2, 0.f); v3 = fmaxf(v3, 0.f);
      } else if (EPI == 2) {
        v0 = v0 > 0.f ? v0 : 0.1f * v0;
        v1 = v1 > 0.f ? v1 : 0.1f * v1;
        v2 = v2 > 0.f ? v2 : 0.1f * v2;
        v3 = v3 > 0.f ? v3 : 0.1f * v3;
      }
      Crow[0] = v0; Crow[16] = v1; Crow[32] = v2; Crow[48] = v3;
    }
  }
}

// CT[z][code] = CB[code][z], c2[code] = sum_z CB[code][z]^2.  1 wave per code.
__global__ __launch_bounds__(256) void codebook_prep(
    const float* __restrict__ CB, float* __restrict__ CT,
    float* __restrict__ c2) {
  const int lane = threadIdx.x & 31;
  const int wave = threadIdx.x >> 5;
  const int code = blockIdx.x * 8 + wave;
  const float* __restrict__ row = CB + (size_t)code * 256;
  float s = 0.f;
#pragma unroll
  for (int z = lane; z < 256; z += 32) {
    const float v = row[z];
    CT[(size_t)z * 1024 + code] = v;
    s += v * v;
  }
#pragma unroll
  for (int off = 16; off > 0; off >>= 1) s += __shfl_xor(s, off, 32);
  if (lane == 0) c2[code] = s;
}

// Nearest code per row: argmin_n (c2[n] - 2 * q.c_n) via WMMA tiles,
// lane-shuffle argmin reduction, then gather codebook rows -> Zemb.
__global__ __launch_bounds__(256) void vq_nearest(
    const float* __restrict__ Q, const float* __restrict__ CT,
    const float* __restrict__ c2, const float* __restrict__ CB,
    float* __restrict__ Zemb) {
  __shared__ int s_idx[128];
  const int lane = threadIdx.x & 31;
  const int wave = threadIdx.x >> 5;
  const int l15  = lane & 15;
  const int half = lane >> 4;
  const int koff = half * 2;
  const int mBase = blockIdx.x * 128 + wave * 16;
  const float* __restrict__ Qrow = Q + (size_t)(mBase + l15) * 256;

  float bestV[8];
  int bestI[8];
#pragma unroll
  for (int r = 0; r < 8; ++r) { bestV[r] = 3.402823466e38f; bestI[r] = 0; }

  for (int ng = 0; ng < 1024; ng += 64) {
    v8f acc0 = {}, acc1 = {}, acc2 = {}, acc3 = {};
    const int nCol = ng + l15;
#pragma unroll 4
    for (int kb = 0; kb < 256; kb += 4) {
      const int k0 = kb + koff;
      v2f a;
      a.x = Qrow[k0];
      a.y = Qrow[k0 + 1];
      const float* __restrict__ C0 = CT + (size_t)k0 * 1024 + nCol;
      const float* __restrict__ C1 = C0 + 1024;
      v2f b0, b1, b2, b3;
      b0.x = C0[0];  b0.y = C1[0];
      b1.x = C0[16]; b1.y = C1[16];
      b2.x = C0[32]; b2.y = C1[32];
      b3.x = C0[48]; b3.y = C1[48];
      acc0 = wmma4(a, b0, acc0);
      acc1 = wmma4(a, b1, acc1);
      acc2 = wmma4(a, b2, acc2);
      acc3 = wmma4(a, b3, acc3);
    }
#pragma unroll
    for (int j = 0; j < 4; ++j) {
      const int n = nCol + j * 16;
      const float cc = c2[n];
      const v8f acc = (j == 0) ? acc0 : (j == 1) ? acc1 : (j == 2) ? acc2 : acc3;
#pragma unroll
      for (int r = 0; r < 8; ++r) {
        const float d = cc - 2.0f * acc[r];
        if (d < bestV[r]) { bestV[r] = d; bestI[r] = n; }
      }
    }
  }

  // reduce across the 16 lanes that hold each row (xor within 16-lane halves)
#pragma unroll
  for (int r = 0; r < 8; ++r) {
    float v = bestV[r];
    int idx = bestI[r];
#pragma unroll
    for (int off = 1; off < 16; off <<= 1) {
      const float ov = __shfl_xor(v, off, 32);
      const int oi = __shfl_xor(idx, off, 32);
      if (ov < v || (ov == v && oi < idx)) { v = ov; idx = oi; }
    }
    if (l15 == 0) s_idx[wave * 16 + half * 8 + r] = idx;
  }
  __syncthreads();

  for (int ml = 0; ml < 16; ++ml) {
    const int idx = s_idx[wave * 16 + ml];
    const float* __restrict__ src = CB + (size_t)idx * 256;
    float* __restrict__ dst = Zemb + (size_t)(mBase + ml) * 256;
    for (int z = lane; z < 256; z += 32) dst[z] = src[z];
  }
}

extern "C" void kernel_launch(void* const* d_in, const int* in_sizes, int n_in,
                              void* d_out, int out_size, void* d_ws,
                              size_t ws_size, hipStream_t stream) {
  (void)in_sizes; (void)n_in; (void)out_size; (void)ws_size;
  const float* X  = (const float*)d_in[0];
  const float* W1 = (const float*)d_in[1];
  const float* b1 = (const float*)d_in[2];
  const float* W2 = (const float*)d_in[3];
  const float* b2 = (const float*)d_in[4];
  const float* CB = (const float*)d_in[5];
  const float* W3 = (const float*)d_in[6];
  const float* b3 = (const float*)d_in[7];
  const float* W4 = (const float*)d_in[8];
  const float* b4 = (const float*)d_in[9];

  const int M = 32 * 512;  // 16384

  float* Zrecon = (float*)d_out;                      // [M,1024]
  float* Zenc   = Zrecon + (size_t)M * 1024;          // [M,256]
  float* Zemb   = Zenc + (size_t)M * 256;             // [M,256]

  float* h  = (float*)d_ws;                           // [M,2048] (reused as h2)
  float* CT = h + (size_t)M * 2048;                   // [256,1024]
  float* c2 = CT + (size_t)256 * 1024;                // [1024]

  const dim3 blk(256);

  codebook_prep<<<1024 / 8, blk, 0, stream>>>(CB, CT, c2);

  // h = relu(X @ W1 + b1)        [M,1024]x[1024,2048]
  wmma_gemm_f32<1><<<dim3(2048 / 64, M / 256), blk, 0, stream>>>(X, W1, b1, h,
                                                                 1024, 2048);
  // Z_enc = h @ W2 + b2          [M,2048]x[2048,256]
  wmma_gemm_f32<0><<<dim3(256 / 64, M / 256), blk, 0, stream>>>(h, W2, b2,
                                                                Zenc, 2048, 256);
  // Z_emb = codebook[argmin dist]
  vq_nearest<<<M / 128, blk, 0, stream>>>(Zenc, CT, c2, CB, Zemb);
  // h2 = leaky_relu(Z_emb @ W3 + b3)   [M,256]x[256,2048]   (reuses h buffer)
  wmma_gemm_f32<2><<<dim3(2048 / 64, M / 256), blk, 0, stream>>>(Zemb, W3, b3,
                                                                 h, 256, 2048);
  // Z_recon = h2 @ W4 + b4       [M,2048]x[2048,1024]
  wmma_gemm_f32<0><<<dim3(1024 / 64, M / 256), blk, 0, stream>>>(h, W4, b4,
                                                                 Zrecon, 2048,
                                                                 1024);
}